// _ProposalLayer_6605659701434
// MI455X (gfx1250) — compile-verified
//
#include <hip/hip_runtime.h>
#include <hip/hip_bf16.h>

typedef __attribute__((ext_vector_type(2))) float v2f;
typedef __attribute__((ext_vector_type(8))) float v8f;

#define NPROP 16464   // 588*(1+5+9+13)
#define NSORT 32768   // next pow2 >= NPROP
#define NPRE  2000
#define NPOST 300
#define NWORDS 63     // ceil(2000/32)
#define ROWTILES 125  // 2000/16
#define IOU_TH 0.7f

// ---------------------------------------------------------------------------
// tube index decode: n -> group(td), t (start frame), h, w, a (base anchor)
// ---------------------------------------------------------------------------
__device__ __forceinline__ void decode_n(int n, int& g, int& td, int& T,
                                         int& t, int& h, int& w, int& a) {
  int off;
  if (n < 588)       { g = 0; td = 16; off = 0;    }
  else if (n < 3528) { g = 1; td = 12; off = 588;  }
  else if (n < 8820) { g = 2; td = 8;  off = 3528; }
  else               { g = 3; td = 4;  off = 8820; }
  T = 17 - td;
  int m = n - off;
  t = m / 588;
  int ka = m % 588;
  h = ka / 84;
  int rem = ka % 84;
  w = rem / 12;
  a = rem % 12;
}

// ---------------------------------------------------------------------------
// 1) gather scores, pack sortable u64 keys (pad with 0 == -inf)
// ---------------------------------------------------------------------------
__global__ void scores_pack(const float* __restrict__ s16, const float* __restrict__ s12,
                            const float* __restrict__ s8,  const float* __restrict__ s4,
                            unsigned long long* __restrict__ keys, int B) {
  int gid = blockIdx.x * blockDim.x + threadIdx.x;
  if (gid >= (B << 15)) return;
  int b = gid >> 15;
  int n = gid & (NSORT - 1);
  unsigned long long key = 0ull;
  if (n < NPROP) {
    int g, td, T, t, h, w, a;
    decode_n(n, g, td, T, t, h, w, a);
    const float* sp = (g == 0) ? s16 : (g == 1) ? s12 : (g == 2) ? s8 : s4;
    size_t idx = ((((size_t)b * 24 + 12 + a) * T + t) * 7 + h) * 7 + w;
    unsigned int u = __float_as_uint(sp[idx]);
    u ^= (u & 0x80000000u) ? 0xFFFFFFFFu : 0x80000000u;   // total-order float
    key = ((unsigned long long)u << 32) |
          (unsigned long long)(0xFFFFFFFFu - (unsigned)n); // ties: smaller idx wins
  }
  keys[gid] = key;
}

// ---------------------------------------------------------------------------
// 2) global bitonic sort pass (ascending); one pass per (k,j)
// ---------------------------------------------------------------------------
__global__ void bitonic_pass(unsigned long long* __restrict__ keys, int k, int j, int B) {
  int gid = blockIdx.x * blockDim.x + threadIdx.x;
  if (gid >= (B << 15)) return;
  int b = gid >> 15;
  int i = gid & (NSORT - 1);
  int ixj = i ^ j;
  if (ixj > i) {
    unsigned long long* kb = keys + ((size_t)b << 15);
    unsigned long long x = kb[i], y = kb[ixj];
    bool up = ((i & k) == 0);
    if (up ? (x > y) : (x < y)) { kb[i] = y; kb[ixj] = x; }
  }
}

// ---------------------------------------------------------------------------
// 3) extract top-2000 (rank-descending from tail of ascending sort)
// ---------------------------------------------------------------------------
__global__ void extract_top(const unsigned long long* __restrict__ keys,
                            int* __restrict__ topIdx, float* __restrict__ topScore, int B) {
  int gid = blockIdx.x * blockDim.x + threadIdx.x;
  if (gid >= B * NPRE) return;
  int b = gid / NPRE, k = gid % NPRE;
  unsigned long long key = keys[((size_t)b << 15) + (NSORT - 1 - k)];
  unsigned int idx = 0xFFFFFFFFu - (unsigned int)(key & 0xFFFFFFFFull);
  unsigned int u = (unsigned int)(key >> 32);
  u = (u & 0x80000000u) ? (u ^ 0x80000000u) : ~u;         // unsortable
  topIdx[gid] = (int)idx;
  topScore[gid] = __uint_as_float(u);
}

// ---------------------------------------------------------------------------
// 4) anchors + bbox_transform_inv + clip -> props (B,2000,16,4), tube areas
// ---------------------------------------------------------------------------
__global__ void build_props(const float* __restrict__ b16, const float* __restrict__ b12,
                            const float* __restrict__ b8,  const float* __restrict__ b4,
                            const float* __restrict__ im_info,
                            const int* __restrict__ topIdx,
                            float* __restrict__ props, float* __restrict__ areas, int B) {
  int gid = blockIdx.x * blockDim.x + threadIdx.x;
  if (gid >= B * NPRE) return;
  int b = gid / NPRE;
  int n = topIdx[gid];
  int g, td, T, t, h, w, a;
  decode_n(n, g, td, T, t, h, w, a);

  const float wsArr[3] = {23.f, 16.f, 11.f};
  const float hsArr[3] = {12.f, 16.f, 22.f};
  int ratio = a >> 2, sc = a & 3;
  float scaleV = (float)(2 << sc);                        // 2,4,8,16
  float Wb = wsArr[ratio] * scaleV, Hb = hsArr[ratio] * scaleV;
  float xc0 = w * 16.f + 7.5f, yc0 = h * 16.f + 7.5f;     // anchor center (w-1 conv)
  float ax1 = xc0 - 0.5f * (Wb - 1.f), ay1 = yc0 - 0.5f * (Hb - 1.f);
  float cx = ax1 + 0.5f * Wb, cy = ay1 + 0.5f * Hb;       // transform_inv uses 0.5*w
  float imh = im_info[b * 3 + 0] - 1.f, imw = im_info[b * 3 + 1] - 1.f;

  const float* bb = (g == 0) ? b16 : (g == 1) ? b12 : (g == 2) ? b8 : b4;
  int C = 12 * td * 4;
  size_t cs = (size_t)T * 49;
  float area = 0.f;
  for (int s = 0; s < 16; ++s) {
    float x1, y1, x2, y2;
    if (s >= t && s < t + td) {
      int f = s - t;
      size_t base = ((((size_t)b * C + (a * td + f) * 4) * T + t) * 7 + h) * 7 + w;
      float dx = bb[base], dy = bb[base + cs];
      float dz = bb[base + 2 * cs], dq = bb[base + 3 * cs];
      float pcx = dx * Wb + cx, pcy = dy * Hb + cy;
      float pw = expf(dz) * Wb, ph = expf(dq) * Hb;
      x1 = pcx - 0.5f * pw; y1 = pcy - 0.5f * ph;
      x2 = pcx + 0.5f * pw; y2 = pcy + 0.5f * ph;
    } else {                                              // zero anchor/delta frame
      x1 = 0.f; y1 = 0.f; x2 = 1.f; y2 = 1.f;
    }
    x1 = fminf(fmaxf(x1, 0.f), imw); y1 = fminf(fmaxf(y1, 0.f), imh);
    x2 = fminf(fmaxf(x2, 0.f), imw); y2 = fminf(fmaxf(y2, 0.f), imh);
    float* dst = props + ((size_t)gid * 16 + s) * 4;
    dst[0] = x1; dst[1] = y1; dst[2] = x2; dst[3] = y2;
    area += (x2 - x1 + 1.f) * (y2 - y1 + 1.f);
  }
  areas[gid] = area;
}

// ---------------------------------------------------------------------------
// 5) suppression bitmask: one wave per 16-row x 32-col tile.
//    Frame-outer / row-inner VALU intersection (only inter[8] stays live).
//    Union's area_i+area_j via V_WMMA_F32_16X16X4_F32 rank-2 outer-sum whose
//    C/D lane layout matches the per-lane inter[] accumulators. Divide-free
//    threshold test. Ballots are wave-uniform: each lane selects its own
//    row's 16 bits in registers (branch-free, no LDS scatter).
// ---------------------------------------------------------------------------
__global__ __launch_bounds__(32) void mask_kernel(const float* __restrict__ props,
                                                  const float* __restrict__ areas,
                                                  unsigned int* __restrict__ mask) {
  int wordId = blockIdx.x;            // 0..62  -> cols [32w, 32w+32)
  int rowTile = blockIdx.y;           // 0..124 -> rows [16r, 16r+16)
  int b = blockIdx.z;
  int lane = threadIdx.x;
  int r0 = rowTile * 16;
  int c0 = wordId * 32;

  __shared__ float4 sRow[16][16];
  __shared__ float4 sCol[32][16];
  __shared__ float sRowArea[16];
  __shared__ float sColArea[32];

  const float4* p4 = reinterpret_cast<const float4*>(props);
  for (int idx = lane; idx < 256; idx += 32) {
    int r = idx >> 4, s = idx & 15;
    sRow[r][s] = p4[((size_t)(b * NPRE + r0 + r)) * 16 + s];
  }
  for (int idx = lane; idx < 512; idx += 32) {
    int c = idx >> 4, s = idx & 15;
    int col = c0 + c;
    float4 v = make_float4(0.f, 0.f, 0.f, 0.f);
    if (col < NPRE) v = p4[((size_t)(b * NPRE + col)) * 16 + s];
    sCol[c][s] = v;
  }
  if (lane < 16) sRowArea[lane] = areas[(size_t)b * NPRE + r0 + lane];
  {
    int col = c0 + lane;
    sColArea[lane] = (col < NPRE) ? areas[(size_t)b * NPRE + col] : 16.f;
  }
  __syncthreads();

  int colLoc = lane & 15;
  bool loHalf = (lane < 16);
  int rowBase = loHalf ? 0 : 8;
  int myK = lane & 7;                 // ballot k that carries this lane's row
  bool myHi = (lane & 8) != 0;        // row in high half of that ballot
  unsigned int word = 0u;             // mask word for row (lane&15), lanes 0-15

  for (int half = 0; half < 2; ++half) {
    int cIdx = half * 16 + colLoc;
    bool colValid = (c0 + cIdx) < NPRE;
    float inter[8] = {0.f, 0.f, 0.f, 0.f, 0.f, 0.f, 0.f, 0.f};
#pragma unroll 2
    for (int s = 0; s < 16; ++s) {
      float4 cb = sCol[cIdx][s];
#pragma unroll
      for (int k = 0; k < 8; ++k) {
        float4 rb = sRow[rowBase + k][s];
        float iw = fmaxf(fminf(rb.z, cb.z) - fmaxf(rb.x, cb.x) + 1.f, 0.f);
        float ih = fmaxf(fminf(rb.w, cb.w) - fmaxf(rb.y, cb.y) + 1.f, 0.f);
        inter[k] += iw * ih;
      }
    }
    // areaSum(i,j) = area_i*1 + 1*area_j : A(16x4) rows [ai,1,0,0], B(4x16)
    // cols [1,aj,0,0]^T. Lanes 0-15 carry K=0..1 slices, lanes 16-31 K=2..3 (=0).
    v2f av, bv;
    av.x = loHalf ? sRowArea[colLoc] : 0.f;
    av.y = loHalf ? 1.f : 0.f;
    bv.x = loHalf ? 1.f : 0.f;
    bv.y = loHalf ? sColArea[half * 16 + colLoc] : 0.f;
    v8f c = {};
    c = __builtin_amdgcn_wmma_f32_16x16x4_f32(false, av, false, bv,
                                              (short)0, c, false, false);
#pragma unroll
    for (int k = 0; k < 8; ++k) {
      // iou > TH  <=>  inter > TH * (areaSum - inter), union strictly > 0
      float uni = c[k] - inter[k];
      bool bit = colValid && (inter[k] > IOU_TH * uni);
      unsigned int bal = (unsigned int)__ballot(bit ? 1 : 0);
      // bal[15:0] = row k, bal[31:16] = row k+8 (cols c0+half*16 .. +15).
      // Branch-free: lane owning row (lane&15) selects its bits itself.
      unsigned int bits = myHi ? (bal >> 16) : (bal & 0xFFFFu);
      word |= ((k == myK) ? bits : 0u) << (half * 16);
    }
  }
  if (lane < 16)
    mask[((size_t)(b * NPRE) + r0 + lane) * NWORDS + wordId] = word;
}

// ---------------------------------------------------------------------------
// 6) greedy NMS reduction + output assembly (one block per batch)
// ---------------------------------------------------------------------------
__global__ __launch_bounds__(64) void nms_kernel(const unsigned int* __restrict__ mask,
                                                 const float* __restrict__ props,
                                                 const float* __restrict__ topScore,
                                                 float* __restrict__ out) {
  int b = blockIdx.x;
  int tid = threadIdx.x;
  __shared__ unsigned int rem[NWORDS];
  __shared__ int keptIdx[NPOST];
  __shared__ int sKeep;
  __shared__ int sCount;
  if (tid < NWORDS) rem[tid] = 0u;
  __syncthreads();
  for (int i = 0; i < NPRE; ++i) {
    if (tid == 0) sKeep = (((rem[i >> 5] >> (i & 31)) & 1u) == 0u) ? 1 : 0;
    __syncthreads();
    if (sKeep && tid < NWORDS) {
      unsigned int m = mask[((size_t)(b * NPRE) + i) * NWORDS + tid];
      int wi = i >> 5;
      if (tid < wi) m = 0u;
      else if (tid == wi) m &= (0xFFFFFFFEu << (i & 31));  // only j > i
      rem[tid] |= m;
    }
    __syncthreads();
  }
  if (tid == 0) {
    int cnt = 0;
    for (int i = 0; i < NPRE; ++i) {
      if (((rem[i >> 5] >> (i & 31)) & 1u) == 0u) {
        if (cnt < NPOST) keptIdx[cnt] = i;
        cnt++;
      }
    }
    sCount = (cnt < NPOST) ? cnt : NPOST;
  }
  __syncthreads();
  for (int p = tid; p < NPOST; p += 64) {
    size_t o = ((size_t)b * NPOST + p) * 66;
    out[o] = (float)b;                                     // col0 always = b
    if (p < sCount) {
      int i = keptIdx[p];
      const float* pp = props + ((size_t)(b * NPRE) + i) * 64;
      for (int q = 0; q < 64; ++q) out[o + 1 + q] = pp[q];
      out[o + 65] = topScore[(size_t)b * NPRE + i];
    } else {
      for (int q = 1; q < 66; ++q) out[o + q] = 0.f;
    }
  }
}

// ---------------------------------------------------------------------------
extern "C" void kernel_launch(void* const* d_in, const int* in_sizes, int n_in,
                              void* d_out, int out_size, void* d_ws, size_t ws_size,
                              hipStream_t stream) {
  const float* s16 = (const float*)d_in[0];
  const float* s12 = (const float*)d_in[1];
  const float* s8  = (const float*)d_in[2];
  const float* s4  = (const float*)d_in[3];
  const float* b16 = (const float*)d_in[4];
  const float* b12 = (const float*)d_in[5];
  const float* b8  = (const float*)d_in[6];
  const float* b4  = (const float*)d_in[7];
  const float* im_info = (const float*)d_in[8];
  float* out = (float*)d_out;
  const int B = in_sizes[8] / 3;

  // workspace carve-up (256B aligned)
  auto alignUp = [](size_t x) { return (x + 255) & ~(size_t)255; };
  char* ws = (char*)d_ws;
  size_t off = 0;
  unsigned long long* keys = (unsigned long long*)(ws + off);
  off = alignUp(off + (size_t)B * NSORT * 8);
  int* topIdx = (int*)(ws + off);
  off = alignUp(off + (size_t)B * NPRE * 4);
  float* topScore = (float*)(ws + off);
  off = alignUp(off + (size_t)B * NPRE * 4);
  float* props = (float*)(ws + off);
  off = alignUp(off + (size_t)B * NPRE * 64 * 4);
  float* areas = (float*)(ws + off);
  off = alignUp(off + (size_t)B * NPRE * 4);
  unsigned int* mask = (unsigned int*)(ws + off);

  int tSort = B * NSORT;
  scores_pack<<<(tSort + 255) / 256, 256, 0, stream>>>(s16, s12, s8, s4, keys, B);
  for (int k = 2; k <= NSORT; k <<= 1)
    for (int j = k >> 1; j > 0; j >>= 1)
      bitonic_pass<<<(tSort + 255) / 256, 256, 0, stream>>>(keys, k, j, B);

  int tTop = B * NPRE;
  extract_top<<<(tTop + 255) / 256, 256, 0, stream>>>(keys, topIdx, topScore, B);
  build_props<<<(tTop + 255) / 256, 256, 0, stream>>>(b16, b12, b8, b4, im_info,
                                                      topIdx, props, areas, B);
  mask_kernel<<<dim3(NWORDS, ROWTILES, B), 32, 0, stream>>>(props, areas, mask);
  nms_kernel<<<B, 64, 0, stream>>>(mask, props, topScore, out);
}